// BlockLinear_MixerBlock_27504970564316
// MI455X (gfx1250) — compile-verified
//
#include <hip/hip_runtime.h>

// ---------------------------------------------------------------------------
// 6-layer radix-4 butterfly mixer, fused into 3 radix-16 stages executed in a
// single HBM pass per 16-row tile held in CDNA5's 320KB LDS.
// Matrix math: V_WMMA_F32_16X16X4_F32 (fp32-exact), 4 chained per 16x16 GEMM.
// Traffic: 256MB (read+write x) + ~0.8MB tables (L2-resident) ~= HBM roofline.
// ---------------------------------------------------------------------------

typedef __attribute__((ext_vector_type(2))) float v2f;
typedef __attribute__((ext_vector_type(8))) float v8f;

#define DIM       4096
#define NBLK      1024
#define ROWS      16
#define LDSTRIDE  4164              // dwords; 4164 % 64 == 4 -> conflict-free row stride
#define LDS_BYTES (ROWS * LDSTRIDE * 4)   // 266,496 B  (< 320KB WGP LDS)

// column skew: stride-256 accesses (stage 2) land on distinct banks,
// preserves float4 alignment for the bulk load/store phases.
__device__ __forceinline__ int pcol(int c) { return c + ((c >> 8) << 2); }

// ---------------------------------------------------------------------------
// Build the 3 x 256 combined 16x16 matrices, stored directly in WMMA-B
// fragment order: [stage][group][q][lane][2] where entry (q,lane,p) holds
// M[j_in = 4q + 2*(lane>>4) + p][j_out = lane&15].
// Derivation (j = h*4 + l within a stride-g group of 16):
//   M[h*4+l_in][m*4+l] = W_A[nA(h)][l_in][l] * W_B[nB(l)][h][m]
//   nA = (stuff*4 + h)*g + t0 ,  nB = stuff*4g + l*g + t0
// ---------------------------------------------------------------------------
__global__ void build_tables_kernel(const float* __restrict__ w,
                                    float* __restrict__ tbl) {
  const int b     = blockIdx.x;          // 0..767  (stage*256 + group)
  const int stage = b >> 8;
  const int gid   = b & 255;
  const int t     = threadIdx.x;         // 256 threads: t = q*64 + L*2 + p
  const int q     = t >> 6;
  const int L     = (t >> 1) & 31;
  const int p     = t & 1;

  const int j_in  = 4 * q + 2 * (L >> 4) + p;
  const int j_out = L & 15;
  const int h  = j_in >> 2,  l_in = j_in & 3;
  const int m2 = j_out >> 2, l    = j_out & 3;

  int g, stuff, t0;
  if (stage == 0)      { g = 1;   stuff = gid;      t0 = 0;        }
  else if (stage == 1) { g = 16;  stuff = gid >> 4; t0 = gid & 15; }
  else                 { g = 256; stuff = 0;        t0 = gid;      }

  const int nA = (stuff * 4 + h) * g + t0;
  const int nB = stuff * 4 * g + l * g + t0;
  const int LA = 2 * stage, LB = LA + 1;

  const float wa = w[((LA * NBLK + nA) * 4 + l_in) * 4 + l];
  const float wb = w[((LB * NBLK + nB) * 4 + h) * 4 + m2];
  tbl[(size_t)b * 256 + t] = wa * wb;
}

// ---------------------------------------------------------------------------
// One workgroup = one 16-row tile, fully resident in LDS. 512 threads = 16
// waves; per stage each wave owns 16 disjoint column-groups (gid = wave+16k),
// so in-place LDS update is race-free; __syncthreads between stages.
// Per group: 4x V_WMMA_F32_16X16X4_F32 accumulate C[16x16] = A[16x16]*B[16x16].
// ---------------------------------------------------------------------------
__global__ void fused_butterfly_kernel(const float* __restrict__ x,
                                       const float* __restrict__ tbl,
                                       float* __restrict__ out) {
  extern __shared__ float tile[];
  const int tid  = threadIdx.x;          // 512
  const int wave = tid >> 5;
  const int lane = tid & 31;
  const size_t rowbase = (size_t)blockIdx.x * ROWS * DIM;

  // ---- bulk load: 16 x 4096 fp32 as float4 (vectorized global reads) ----
  for (int i = tid; i < ROWS * (DIM / 4); i += 512) {
    const int r  = i >> 10;              // 1024 float4 per row
    const int c4 = i & 1023;
    const float4 v = reinterpret_cast<const float4*>(x + rowbase + (size_t)r * DIM)[c4];
    *reinterpret_cast<float4*>(&tile[r * LDSTRIDE + pcol(c4 * 4)]) = v;
  }
  __syncthreads();

  // A-fragment addressing (16x4 fp32): lanes 0-15 hold K=0,1; 16-31 hold K=2,3
  const int Arow = lane & 15;
  const int kk2  = (lane >> 4) << 1;

  #pragma unroll
  for (int stage = 0; stage < 3; ++stage) {
    const int g = (stage == 0) ? 1 : (stage == 1) ? 16 : 256;
    for (int gid = wave; gid < 256; gid += 16) {
      int base;                                     // col = base + j*g
      if (stage == 0)      base = gid << 4;
      else if (stage == 1) base = ((gid >> 4) << 8) + (gid & 15);
      else                 base = gid;

      const v2f* __restrict__ B =
          reinterpret_cast<const v2f*>(tbl + (((size_t)stage * 256 + gid) << 8));

      v8f c = {};
      #pragma unroll
      for (int q = 0; q < 4; ++q) {
        v2f a;
        const int j0 = 4 * q + kk2;
        a.x = tile[Arow * LDSTRIDE + pcol(base + j0 * g)];
        a.y = tile[Arow * LDSTRIDE + pcol(base + (j0 + 1) * g)];
        const v2f bb = B[q * 32 + lane];           // pre-swizzled B fragment
        // D = A(16x4) * B(4x16) + C   -> v_wmma_f32_16x16x4_f32
        c = __builtin_amdgcn_wmma_f32_16x16x4_f32(
                false, a, false, bb, (short)0, c, false, false);
      }

      // C/D layout: VGPR v -> row (v + 8*(lane>>4)), col j_out = lane&15
      const int ccol = pcol(base + (lane & 15) * g);
      const int radd = (lane >> 4) << 3;
      #pragma unroll
      for (int v = 0; v < 8; ++v)
        tile[(v + radd) * LDSTRIDE + ccol] = c[v];
    }
    __syncthreads();                               // stages mix across groups
  }

  // ---- bulk store ----
  for (int i = tid; i < ROWS * (DIM / 4); i += 512) {
    const int r  = i >> 10;
    const int c4 = i & 1023;
    const float4 v = *reinterpret_cast<const float4*>(&tile[r * LDSTRIDE + pcol(c4 * 4)]);
    reinterpret_cast<float4*>(out + rowbase + (size_t)r * DIM)[c4] = v;
  }
}

// ---------------------------------------------------------------------------
extern "C" void kernel_launch(void* const* d_in, const int* in_sizes, int n_in,
                              void* d_out, int out_size, void* d_ws, size_t ws_size,
                              hipStream_t stream) {
  const float* x = (const float*)d_in[0];   // [8192, 4096] fp32
  const float* w = (const float*)d_in[1];   // [6, 1024, 4, 4] fp32
  float* out = (float*)d_out;               // [8192, 4096] fp32
  float* tbl = (float*)d_ws;                // needs 3*256*256*4 = 786,432 B

  // opt-in to >64KB dynamic LDS (idempotent; not a stream op, capture-safe)
  (void)hipFuncSetAttribute((const void*)fused_butterfly_kernel,
                            hipFuncAttributeMaxDynamicSharedMemorySize,
                            LDS_BYTES);

  build_tables_kernel<<<768, 256, 0, stream>>>(w, tbl);
  fused_butterfly_kernel<<<8192 / ROWS, 512, LDS_BYTES, stream>>>(x, tbl, out);
}